// SFCPointTokenizer_19172734009550
// MI455X (gfx1250) — compile-verified
//
#include <hip/hip_runtime.h>
#include <hip/hip_bf16.h>
#include <stdint.h>

typedef __attribute__((ext_vector_type(2))) float v2f;
typedef __attribute__((ext_vector_type(8))) float v8f;

#define NBATCH 16
#define NPTS   4096
#define NDOUT  384
#define HB     8

// Exact port of the Skilling transform in the reference (n=3 dims, p=8 bits).
__device__ __forceinline__ int hilbert3(int x0, int x1, int x2) {
  const int m = 1 << (HB - 1);
  for (int q = m; q > 1; q >>= 1) {
    int pm = q - 1;
    // i = 0
    x0 = (x0 & q) ? (x0 ^ pm) : x0;
    // i = 1
    {
      int t  = (x0 ^ x1) & pm;
      bool c = (x1 & q) != 0;
      int x0n = c ? (x0 ^ pm) : (x0 ^ t);
      int x1n = c ? x1 : (x1 ^ t);
      x0 = x0n; x1 = x1n;
    }
    // i = 2
    {
      int t  = (x0 ^ x2) & pm;
      bool c = (x2 & q) != 0;
      int x0n = c ? (x0 ^ pm) : (x0 ^ t);
      int x2n = c ? x2 : (x2 ^ t);
      x0 = x0n; x2 = x2n;
    }
  }
  // Gray encode
  x1 ^= x0;
  x2 ^= x1;
  int t = 0;
  for (int q = m; q > 1; q >>= 1)
    t = (x2 & q) ? (t ^ (q - 1)) : t;
  x0 ^= t; x1 ^= t; x2 ^= t;
  // Interleave bits MSB-first, dim order 0,1,2 -> 24-bit index
  int h = 0;
  for (int bp = HB - 1; bp >= 0; --bp) {
    h = (h << 1) | ((x0 >> bp) & 1);
    h = (h << 1) | ((x1 >> bp) & 1);
    h = (h << 1) | ((x2 >> bp) & 1);
  }
  return h;
}

// K1: per-batch min/max reduction + Hilbert codes for both coordinate orders.
// key = (hilbert << 12) | point_index   (stable argsort via packed tie-break)
__global__ void __launch_bounds__(256) sfc_codes_kernel(
    const float* __restrict__ x, unsigned long long* __restrict__ keys) {
  __shared__ float smn[3][256];
  __shared__ float smx[3][256];
  const int b   = blockIdx.x;
  const int tid = threadIdx.x;
  const float* xb = x + (size_t)b * NPTS * 3;

  float mn0 =  3.4e38f, mn1 =  3.4e38f, mn2 =  3.4e38f;
  float mx0 = -3.4e38f, mx1 = -3.4e38f, mx2 = -3.4e38f;
  for (int n = tid; n < NPTS; n += 256) {
    float c0 = xb[n * 3 + 0], c1 = xb[n * 3 + 1], c2 = xb[n * 3 + 2];
    mn0 = fminf(mn0, c0); mx0 = fmaxf(mx0, c0);
    mn1 = fminf(mn1, c1); mx1 = fmaxf(mx1, c1);
    mn2 = fminf(mn2, c2); mx2 = fmaxf(mx2, c2);
  }
  smn[0][tid] = mn0; smn[1][tid] = mn1; smn[2][tid] = mn2;
  smx[0][tid] = mx0; smx[1][tid] = mx1; smx[2][tid] = mx2;
  __syncthreads();
  for (int s = 128; s > 0; s >>= 1) {
    if (tid < s) {
      #pragma unroll
      for (int c = 0; c < 3; ++c) {
        smn[c][tid] = fminf(smn[c][tid], smn[c][tid + s]);
        smx[c][tid] = fmaxf(smx[c][tid], smx[c][tid + s]);
      }
    }
    __syncthreads();
  }
  const float m0 = smn[0][0], m1 = smn[1][0], m2 = smn[2][0];
  const float s0 = fmaxf(smx[0][0] - m0, 1e-6f);
  const float s1 = fmaxf(smx[1][0] - m1, 1e-6f);
  const float s2 = fmaxf(smx[2][0] - m2, 1e-6f);

  unsigned long long* kh = keys + ((size_t)b * 2 + 0) * NPTS;
  unsigned long long* kt = keys + ((size_t)b * 2 + 1) * NPTS;
  for (int n = tid; n < NPTS; n += 256) {
    float c0 = xb[n * 3 + 0], c1 = xb[n * 3 + 1], c2 = xb[n * 3 + 2];
    int g0 = (int)fminf(fmaxf((c0 - m0) / s0 * 255.0f, 0.0f), 255.0f);
    int g1 = (int)fminf(fmaxf((c1 - m1) / s1 * 255.0f, 0.0f), 255.0f);
    int g2 = (int)fminf(fmaxf((c2 - m2) / s2 * 255.0f, 0.0f), 255.0f);
    int h  = hilbert3(g0, g1, g2);   // xyz order
    int ht = hilbert3(g2, g1, g0);   // zyx order (norm[..., ::-1])
    kh[n] = ((unsigned long long)(unsigned)h  << 12) | (unsigned)n;
    kt[n] = ((unsigned long long)(unsigned)ht << 12) | (unsigned)n;
  }
}

// K2: in-LDS bitonic sort of 4096 packed uint64 keys (32 KB of the 320 KB WGP LDS).
__global__ void __launch_bounds__(512) sfc_sort_kernel(
    unsigned long long* __restrict__ keys) {
  __shared__ unsigned long long s[NPTS];
  unsigned long long* kb = keys + (size_t)blockIdx.x * NPTS;
  const int tid = threadIdx.x;
  for (int i = tid; i < NPTS; i += 512) s[i] = kb[i];
  __syncthreads();
  for (int k = 2; k <= NPTS; k <<= 1) {
    for (int j = k >> 1; j > 0; j >>= 1) {
      for (int i = tid; i < NPTS; i += 512) {
        int ij = i ^ j;
        if (ij > i) {
          unsigned long long a = s[i], c = s[ij];
          bool up = (i & k) == 0;
          if ((a > c) == up) { s[i] = c; s[ij] = a; }
        }
      }
      __syncthreads();
    }
  }
  for (int i = tid; i < NPTS; i += 512) kb[i] = s[i];
}

// K3: gather + Linear(3->384) via V_WMMA_F32_16X16X4_F32 + gamma/beta, store-bound.
// Per block (8 waves): stage B-fragments (W with K-pad baked in) and fused
// {scale, shift} per channel in LDS once; steady loop is ds_load_b64 x2 ->
// WMMA -> 8 FMA -> 8-store clause. No global loads, no EXEC divergence.
// A frag (16x4 f32): lanes 0-15 hold K=0,1 ; lanes 16-31 hold K=2,K=3.
//   A's K=3 slot may hold garbage because B's K=3 row is zeroed in LDS.
// D (16x16 f32): VGPR j -> row j (lanes 0-15) / row j+8 (lanes 16-31).
__global__ void __launch_bounds__(256) sfc_tokenize_kernel(
    const float* __restrict__ x, const float* __restrict__ W,
    const float* __restrict__ bias, const float* __restrict__ gamma,
    const float* __restrict__ beta,
    const unsigned long long* __restrict__ keys,
    float* __restrict__ out) {
  __shared__ v2f sB[24 * 32];   // B fragments per dtile per lane (6 KB)
  __shared__ v2f sS[NDOUT];     // {scale, shift} per channel      (3 KB)

  const int tileBlk = blockIdx.x;   // 0..31 (token super-tile)
  const int v       = blockIdx.y;   // 0..1  (variant h / t)
  const int bb      = blockIdx.z;   // 0..15 (batch)
  const int tid  = threadIdx.x;
  const int wave = tid >> 5;
  const int lane = tid & 31;
  const int row  = lane & 15;
  const bool lo  = lane < 16;

  // ---- stage parameters in LDS (once per block) ----
  for (int idx = tid; idx < 24 * 32; idx += 256) {
    int dts = idx >> 5, ln = idx & 31, n = dts * 16 + (ln & 15);
    const float* wr = W + n * 3;
    v2f bf;
    if (ln < 16) { bf.x = wr[0]; bf.y = wr[1]; }   // K=0, K=1
    else         { bf.x = wr[2]; bf.y = 0.0f; }    // K=2, K=3 (zero pad)
    sB[idx] = bf;
  }
  for (int n = tid; n < NDOUT; n += 256) {
    float sc = gamma[v * NDOUT + n];
    v2f ss; ss.x = sc; ss.y = fmaf(bias[n], sc, beta[v * NDOUT + n]);
    sS[n] = ss;
  }
  __syncthreads();

  // ---- build A fragment from the gathered point ----
  const int tileBase = (tileBlk * 8 + wave) * 16;   // first token of this wave's tile
  const unsigned long long* kb = keys + ((size_t)bb * 2 + v) * NPTS;
  const int ord = (int)(kb[tileBase + row] & 0xFFFULL);
  const float* p = x + ((size_t)bb * NPTS + ord) * 3;
  v2f afrag;
  afrag.x = p[lo ? 0 : 2];   // K=0 (lo) / K=2 (hi), address select: no exec divergence
  afrag.y = p[1];            // K=1 (lo); K=3 slot for hi lanes is nulled by B's zero row

  const size_t rowBase = (size_t)bb * (2 * NPTS) + (size_t)v * NPTS + (size_t)tileBase
                       + (size_t)(lo ? 0 : 8);
  float* obase = out + rowBase * NDOUT + row;

  #pragma unroll 2
  for (int dt = 0; dt < 24; ++dt) {
    const int n = dt * 16 + row;                // output channel for this lane
    v2f bfrag = sB[dt * 32 + lane];
    v2f ss    = sS[n];

    v8f c = {};
    c = __builtin_amdgcn_wmma_f32_16x16x4_f32(false, afrag, false, bfrag,
                                              (short)0, c, false, false);

    float* o = obase + dt * 16;
    #pragma unroll
    for (int j = 0; j < 8; ++j)
      o[(size_t)j * NDOUT] = fmaf(c[j], ss.x, ss.y);
  }
}

extern "C" void kernel_launch(void* const* d_in, const int* in_sizes, int n_in,
                              void* d_out, int out_size, void* d_ws, size_t ws_size,
                              hipStream_t stream) {
  (void)in_sizes; (void)n_in; (void)out_size; (void)ws_size;
  const float* x     = (const float*)d_in[0];   // [16,4096,3]
  const float* W     = (const float*)d_in[1];   // [384,3]
  const float* bias  = (const float*)d_in[2];   // [384]
  const float* gamma = (const float*)d_in[3];   // [2,384]
  const float* beta  = (const float*)d_in[4];   // [2,384]
  float* out = (float*)d_out;                   // [16,8192,384]
  unsigned long long* keys = (unsigned long long*)d_ws;  // [16][2][4096] packed keys (1 MB)

  sfc_codes_kernel<<<NBATCH, 256, 0, stream>>>(x, keys);
  sfc_sort_kernel<<<NBATCH * 2, 512, 0, stream>>>(keys);
  dim3 grid(NPTS / 128, 2, NBATCH);             // 32 x 2 x 16 = 1024 blocks, 8 waves each
  sfc_tokenize_kernel<<<grid, 256, 0, stream>>>(x, W, bias, gamma, beta, keys, out);
}